// DeepFace_10574209482846
// MI455X (gfx1250) — compile-verified
//
#include <hip/hip_runtime.h>
#include <math.h>
#include <stdint.h>

typedef float v2f __attribute__((ext_vector_type(2)));
typedef float v8f __attribute__((ext_vector_type(8)));
typedef int   v4i_vs __attribute__((vector_size(16)));  // matches builtin param

#define BATCH 2048

#if __has_builtin(__builtin_amdgcn_global_load_async_to_lds_b128) && \
    __has_builtin(__builtin_amdgcn_s_wait_asynccnt)
#define HAS_ASYNC_LDS 1
#else
#define HAS_ASYNC_LDS 0
#endif

// Full-precision fp32 WMMA: D(16x16) = A(16x4) * B(4x16) + C, wave32.
__device__ __forceinline__ v8f wmma_f32(v2f a, v2f b, v8f c) {
  return __builtin_amdgcn_wmma_f32_16x16x4_f32(false, a, false, b, (short)0, c,
                                               false, false);
}

// ---------------------------------------------------------------------------
// Repack x: NCHW (B,5,60,60) -> [h][w][c_pad8][b], zero-padding channels 5..7.
// Coalesced stores; whole x (147MB) fits in 192MB L2 so gathers mostly hit L2.
// ---------------------------------------------------------------------------
__global__ void repack_x(const float* __restrict__ x, float* __restrict__ xp) {
  unsigned tid = blockIdx.x * blockDim.x + threadIdx.x;  // total 60*60*8*2048
  unsigned b = tid & (BATCH - 1);
  unsigned r = tid / BATCH;
  unsigned c = r & 7; r >>= 3;
  unsigned w = r % 60, h = r / 60;
  float v = 0.f;
  if (c < 5) v = x[(((size_t)b * 5 + c) * 60 + h) * 60 + w];
  xp[tid] = v;
}

// ---------------------------------------------------------------------------
// Repack weights (loc,co,ci,tap) -> [loc][tap][ci_pad][co], zero pad ci.
// Handles tied (nloc=1) and locally-connected weights uniformly.
// ---------------------------------------------------------------------------
__global__ void repack_w(const float* __restrict__ in, float* __restrict__ outw,
                         int nloc, int ntap, int Ci, int CiPad, int Co) {
  unsigned tid = blockIdx.x * blockDim.x + threadIdx.x;
  unsigned total = (unsigned)nloc * ntap * CiPad * Co;
  if (tid >= total) return;
  unsigned co = tid % Co;  unsigned r = tid / Co;
  unsigned ci = r % CiPad; r /= CiPad;
  unsigned tap = r % ntap; unsigned loc = r / ntap;
  float v = 0.f;
  if ((int)ci < Ci)
    v = in[(((size_t)loc * Co + co) * Ci + ci) * ntap + tap];
  outw[tid] = v;
}

// ---------------------------------------------------------------------------
// Implicit-GEMM conv via V_WMMA_F32_16X16X4_F32.
//   in : [Hi][Wi][CI][B]   wgt: [loc][tap][CI][Co]   out: [Ho][Wo][Co][B]
// Grid: (Wo, 16 batch groups, Ho); block = 128 threads = 4 waves.
// Each wave: 32(batch) x Co tile (two A frags, 2*NT accumulators).
// Weights for TC taps staged in LDS (async global->LDS when available),
// per-ci row padded to Co+16 floats so the two half-wave B reads hit
// disjoint bank ranges (diff = 2*(Co+16) = 160 ≡ 32 mod 64).
// A frag: lane l, vgpr v -> (b = b0 + l%16,        k = base + v + 2*(l>>4))
// B frag: lane l, vgpr v -> (k = base + v+2*(l>>4), co = co0 + l%16)
// D tile: lane l, vgpr v -> (b = b0 + v + 8*(l>>4), co = l%16) => float4 stores
// ---------------------------------------------------------------------------
template <int NT, int KH, int KW, int STRIDE, int CI, int TC>
__global__ __launch_bounds__(128) void conv_wmma(
    const float* __restrict__ in, const float* __restrict__ wgt,
    const float* __restrict__ bias, float* __restrict__ out,
    int Wi, int wLocStride, int biasLocStride) {
  constexpr int Co    = NT * 16;
  constexpr int CoP   = Co + 16;        // padded row stride in LDS
  constexpr int NTAPS = KH * KW;
  __shared__ float lwgt[TC * CI * CoP];

  const int lane = threadIdx.x & 31;
  const int wave = threadIdx.x >> 5;
  const int l16  = lane & 15;
  const int hl   = lane >> 4;       // half-wave id
  const int koff = hl << 1;         // K offset of this half-wave within a group
  const int wo = blockIdx.x, ho = blockIdx.z;
  const int loc = ho * gridDim.x + wo;
  const int b0 = ((blockIdx.y << 2) + wave) << 5;   // 32 batch rows per wave

  const float* wloc = wgt + (size_t)loc * wLocStride;
  v8f acc0[NT] = {};
  v8f acc1[NT] = {};

  for (int tb = 0; tb < NTAPS; tb += TC) {
    const int tcnt = (NTAPS - tb) < TC ? (NTAPS - tb) : TC;
    // ---- stage weight chunk [tb, tb+tcnt) into LDS (row-padded) ----
    const float* gsrc = wloc + (size_t)tb * CI * Co;
    const int cfG = tcnt * CI * Co;
    for (int i = threadIdx.x * 4; i < cfG; i += 128 * 4) {
      const int row = i / Co;                 // Co is a power-of-two constant
      const int col = i - row * Co;
      float* dst = &lwgt[row * CoP + col];
#if HAS_ASYNC_LDS
      __builtin_amdgcn_global_load_async_to_lds_b128(
          (v4i_vs*)(gsrc + i), (v4i_vs*)dst, 0, 0);
#else
      *(float4*)dst = *(const float4*)(gsrc + i);
#endif
    }
#if HAS_ASYNC_LDS
    __builtin_amdgcn_s_wait_asynccnt(0);
#endif
    __syncthreads();

    // ---- compute over the staged taps ----
    #pragma unroll 1
    for (int t = 0; t < tcnt; ++t) {
      const int tap = tb + t;
      const int kh = tap / KW, kw = tap % KW;  // constant divisor -> cheap
      const int hi = ho * STRIDE + kh;
      const int wi = wo * STRIDE + kw;
      const float* aB0 = in + (size_t)(hi * Wi + wi) * CI * BATCH + b0 + l16;
      const float* aB1 = aB0 + 16;
      const float* wB  = &lwgt[t * CI * CoP + l16];
      #pragma unroll
      for (int c = 0; c < CI; c += 4) {
        v2f A0, A1;
        A0.x = aB0[(size_t)(c + koff) * BATCH];
        A0.y = aB0[(size_t)(c + koff + 1) * BATCH];
        A1.x = aB1[(size_t)(c + koff) * BATCH];
        A1.y = aB1[(size_t)(c + koff + 1) * BATCH];
        #pragma unroll
        for (int nt = 0; nt < NT; ++nt) {
          v2f Bf;
          Bf.x = wB[(c + koff) * CoP + nt * 16];
          Bf.y = wB[(c + koff + 1) * CoP + nt * 16];
          acc0[nt] = wmma_f32(A0, Bf, acc0[nt]);
          acc1[nt] = wmma_f32(A1, Bf, acc1[nt]);
        }
      }
    }
    __syncthreads();
  }

  // Epilogue: bias + tanh, vectorized 16B stores (8 consecutive b per lane).
  #pragma unroll
  for (int nt = 0; nt < NT; ++nt) {
    const int co = nt * 16 + l16;
    const float bv = bias[(size_t)loc * biasLocStride + co];
    float* opBase = out + ((size_t)loc * Co + co) * BATCH + b0 + (hl << 3);
    #pragma unroll
    for (int m = 0; m < 2; ++m) {
      const v8f& a = m ? acc1[nt] : acc0[nt];
      float4 lo, hi4;
      lo.x  = tanhf(a[0] + bv);
      lo.y  = tanhf(a[1] + bv);
      lo.z  = tanhf(a[2] + bv);
      lo.w  = tanhf(a[3] + bv);
      hi4.x = tanhf(a[4] + bv);
      hi4.y = tanhf(a[5] + bv);
      hi4.z = tanhf(a[6] + bv);
      hi4.w = tanhf(a[7] + bv);
      float* op = opBase + m * 16;
      *(float4*)(op)     = lo;
      *(float4*)(op + 4) = hi4;
    }
  }
}

// ---------------------------------------------------------------------------
// Head: flatten (NCHW order: f = c*9 + loc), concat info, 584->2 linear,
// softmax over 2 logits. One thread per batch element, coalesced in b.
// ---------------------------------------------------------------------------
__global__ void head_kernel(const float* __restrict__ a6,
                            const float* __restrict__ info,
                            const float* __restrict__ hw,
                            const float* __restrict__ hb,
                            float* __restrict__ out) {
  int b = blockIdx.x * blockDim.x + threadIdx.x;  // 2048 threads
  float acc0 = hb[0], acc1 = hb[1];
  for (int c = 0; c < 64; ++c) {
    #pragma unroll
    for (int loc = 0; loc < 9; ++loc) {
      int f = c * 9 + loc;
      float v = a6[((size_t)loc * 64 + c) * BATCH + b];
      acc0 += v * hw[f];
      acc1 += v * hw[584 + f];
    }
  }
  #pragma unroll
  for (int j = 0; j < 8; ++j) {
    float v = info[b * 8 + j];
    acc0 += v * hw[576 + j];
    acc1 += v * hw[584 + 576 + j];
  }
  float m = fmaxf(acc0, acc1);
  float e0 = expf(acc0 - m), e1 = expf(acc1 - m);
  float s = e0 + e1;
  out[b * 2 + 0] = e0 / s;
  out[b * 2 + 1] = e1 / s;
}

// ---------------------------------------------------------------------------
extern "C" void kernel_launch(void* const* d_in, const int* in_sizes, int n_in,
                              void* d_out, int out_size, void* d_ws,
                              size_t ws_size, hipStream_t stream) {
  (void)in_sizes; (void)n_in; (void)out_size; (void)ws_size;
  const float* x    = (const float*)d_in[0];
  const float* info = (const float*)d_in[1];
  const float* w1   = (const float*)d_in[2];
  const float* b1   = (const float*)d_in[3];
  const float* w2a  = (const float*)d_in[4];
  const float* b2a  = (const float*)d_in[5];
  const float* w2b  = (const float*)d_in[6];
  const float* b2b  = (const float*)d_in[7];
  const float* lw3  = (const float*)d_in[8];
  const float* lb3  = (const float*)d_in[9];
  const float* lw4  = (const float*)d_in[10];
  const float* lb4  = (const float*)d_in[11];
  const float* lw5  = (const float*)d_in[12];
  const float* lb5  = (const float*)d_in[13];
  const float* hw   = (const float*)d_in[14];
  const float* hb   = (const float*)d_in[15];
  float* out = (float*)d_out;
  float* ws  = (float*)d_ws;

  // Ping-pong activation regions (float counts):
  //  regA: xp(58.98M) -> a2(95.55M) -> a4(10.62M) -> a6(1.18M)
  //  regB: a1(55.12M) -> a3(22.15M) -> a5(3.28M)
  const size_t REGA = 95551488;  // 27*27*64*2048
  const size_t REGB = 55115776;  // 29*29*32*2048
  float* regA = ws;
  float* regB = ws + REGA;
  float* w1r  = ws + REGA + REGB;      // 9*8*32      = 2304
  float* w2ar = w1r  + 2304;           // 9*32*64     = 18432
  float* w2br = w2ar + 18432;          // 9*64*64     = 36864
  float* lw3r = w2br + 36864;          // 81*25*64*64 = 8294400
  float* lw4r = lw3r + 8294400;        // 25*25*64*64 = 2560000
  float* lw5r = lw4r + 2560000;        // 9*9*64*64   = 331776

  auto cdiv = [](size_t a, size_t b) { return (unsigned)((a + b - 1) / b); };

  // Weight repacks (tiny)
  repack_w<<<cdiv(9 * 8 * 32, 256), 256, 0, stream>>>(w1, w1r, 1, 9, 5, 8, 32);
  repack_w<<<cdiv(9 * 32 * 64, 256), 256, 0, stream>>>(w2a, w2ar, 1, 9, 32, 32, 64);
  repack_w<<<cdiv(9 * 64 * 64, 256), 256, 0, stream>>>(w2b, w2br, 1, 9, 64, 64, 64);
  repack_w<<<cdiv(81ull * 25 * 64 * 64, 256), 256, 0, stream>>>(lw3, lw3r, 81, 25, 64, 64, 64);
  repack_w<<<cdiv(25ull * 25 * 64 * 64, 256), 256, 0, stream>>>(lw4, lw4r, 25, 25, 64, 64, 64);
  repack_w<<<cdiv(9 * 9 * 64 * 64, 256), 256, 0, stream>>>(lw5, lw5r, 9, 9, 64, 64, 64);

  // Input repack NCHW -> [h][w][c8][b]
  repack_x<<<cdiv(60ull * 60 * 8 * BATCH, 256), 256, 0, stream>>>(x, regA);

  dim3 blk(128, 1, 1);  // 4 waves * 32 batch; grid.y=16 -> 2048 batch
  // L1: (B,8,60,60)->(B,32,29,29) k3 s2
  conv_wmma<2, 3, 3, 2, 8, 9><<<dim3(29, 16, 29), blk, 0, stream>>>(
      regA, w1r, b1, regB, 60, 0, 0);
  // L2a: ->(B,64,27,27) k3 s1
  conv_wmma<4, 3, 3, 1, 32, 6><<<dim3(27, 16, 27), blk, 0, stream>>>(
      regB, w2ar, b2a, regA, 29, 0, 0);
  // L2b: ->(B,64,13,13) k3 s2
  conv_wmma<4, 3, 3, 2, 64, 3><<<dim3(13, 16, 13), blk, 0, stream>>>(
      regA, w2br, b2b, regB, 27, 0, 0);
  // L3 (local): ->(B,64,9,9) k5
  conv_wmma<4, 5, 5, 1, 64, 3><<<dim3(9, 16, 9), blk, 0, stream>>>(
      regB, lw3r, lb3, regA, 13, 25 * 64 * 64, 64);
  // L4 (local): ->(B,64,5,5) k5
  conv_wmma<4, 5, 5, 1, 64, 3><<<dim3(5, 16, 5), blk, 0, stream>>>(
      regA, lw4r, lb4, regB, 9, 25 * 64 * 64, 64);
  // L5 (local): ->(B,64,3,3) k3
  conv_wmma<4, 3, 3, 1, 64, 3><<<dim3(3, 16, 3), blk, 0, stream>>>(
      regB, lw5r, lb5, regA, 5, 9 * 64 * 64, 64);
  // Head + softmax
  head_kernel<<<BATCH / 256, 256, 0, stream>>>(regA, info, hw, hb, out);
}